// RetinexLoss_1632087572736
// MI455X (gfx1250) — compile-verified
//
#include <hip/hip_runtime.h>

typedef float v2f __attribute__((ext_vector_type(2)));
typedef float v4f __attribute__((ext_vector_type(4)));
typedef float v8f __attribute__((ext_vector_type(8)));

#define NEL   12582912   // 16*3*512*512
#define NB    16
#define CCH   3
#define HH    512
#define WW    512
#define SC    (HH*WW)      // 262144 channel stride
#define SN    (CCH*SC)     // 786432 batch stride
#define SC4   (SC/4)
#define SN4   (SN/4)

// Gaussian kernel sigma=1, radius=4, normalized -- used ONLY with compile-time
// indices (folds to immediates after unroll).
__device__ __forceinline__ float gk(int i) {
  const float K[9] = {1.338341e-4f, 4.431861e-3f, 5.399097e-2f, 2.419713e-1f,
                      3.989434e-1f, 2.419713e-1f, 5.399097e-2f, 4.431861e-3f,
                      1.338341e-4f};
  return K[i];
}

// Fill the shared 64-entry banded-weight table: tbl[i] = k[i-20] for
// i in [20,28], else 0.  One v_exp_f32 per lane, once per block.
__device__ __forceinline__ void fill_band_table(float* tbl) {
  if (threadIdx.x < 64) {
    float t = (float)((int)threadIdx.x - 24);           // (d - 4), d = i - 20
    float wv = 0.3989434f * __expf(-0.5f * t * t);
    tbl[threadIdx.x] = ((unsigned)(threadIdx.x - 20) <= 8u) ? wv : 0.f;
  }
}

// ---- Pass 1: fused log + blur along W.  16 rows x 16 out-cols per wave.
// out(16x16) = A(16x24 window) x B(24x16 band), 6 x V_WMMA_F32_16X16X4_F32
__global__ void k_wblur_log(const float* __restrict__ in, float* __restrict__ out) {
  __shared__ float sm[8 * 384 + 64];
  float* tbl = &sm[8 * 384];
  const int lane = threadIdx.x & 31;
  const int wv   = threadIdx.x >> 5;
  const int tile = blockIdx.x * 8 + wv;      // 49152 tiles total
  const int rt   = tile >> 5;                // row-tile (1536)
  const int wt   = tile & 31;                // w-tile (32)
  float* w = &sm[wv * 384];
  const long rowbase = (long)rt * 16 * WW;

  fill_band_table(tbl);
  // stage 16x24 window into LDS (symmetric fold on W), log fused
  for (int i = lane; i < 384; i += 32) {     // 12 full-wave iters, no divergence
    int r = i / 24, u = i - r * 24;
    int gw = wt * 16 - 4 + u;
    gw = (gw < 0) ? (-1 - gw) : (gw > 511 ? 1023 - gw : gw);
    float v = in[rowbase + (long)r * WW + gw];
    w[r * 24 + u] = __logf(v + 1e-6f);
  }
  __syncthreads();

  const int m  = lane & 15;
  const int hi = lane >> 4;                  // lane half selects K pair / M+8
  const int bx = hi - m + 20;                // per-lane table base, in [5,21]
  v8f acc = {};
#pragma unroll
  for (int s = 0; s < 6; ++s) {
    const int u0 = 4 * s + (hi << 1);
    v2f a; a.x = w[m * 24 + u0];   a.y = w[m * 24 + u0 + 1];   // A: window
    v2f b; b.x = tbl[bx + 4 * s];  b.y = tbl[bx + 4 * s + 2];  // B: band table
    acc = __builtin_amdgcn_wmma_f32_16x16x4_f32(false, a, false, b,
                                                (short)0, acc, false, false);
  }
#pragma unroll
  for (int v = 0; v < 8; ++v)                // D layout: M = v + 8*hi, N = m
    out[rowbase + (long)(v + hi * 8) * WW + wt * 16 + m] = acc[v];
}

// ---- Pass 2: blur along H.  out(16x16) = A(16x24 band) x B(24x16 window)
__global__ void k_hblur(const float* __restrict__ in, float* __restrict__ out) {
  __shared__ float sm[8 * 384 + 64];
  float* tbl = &sm[8 * 384];
  const int lane = threadIdx.x & 31;
  const int wv   = threadIdx.x >> 5;
  const int tile = blockIdx.x * 8 + wv;      // 49152 = 48 planes * 32 * 32
  const int plane = tile >> 10;
  const int rem   = tile & 1023;
  const int ht = rem >> 5, wt = rem & 31;
  float* w = &sm[wv * 384];
  const long pbase = (long)plane * SC;

  fill_band_table(tbl);
  for (int i = lane; i < 384; i += 32) {     // 24 h-rows x 16 w-cols
    int u = i >> 4, j = i & 15;
    int gh = ht * 16 - 4 + u;
    gh = (gh < 0) ? (-1 - gh) : (gh > 511 ? 1023 - gh : gh);
    w[u * 16 + j] = in[pbase + (long)gh * WW + wt * 16 + j];
  }
  __syncthreads();

  const int m = lane & 15, hi = lane >> 4;
  const int ax = (hi << 1) - m + 20;         // per-lane table base, in [5,22]
  v8f acc = {};
#pragma unroll
  for (int s = 0; s < 6; ++s) {
    v2f a; a.x = tbl[ax + 4 * s];            a.y = tbl[ax + 4 * s + 1]; // A: band
    v2f b; b.x = w[(4 * s + hi) * 16 + m];   b.y = w[(4 * s + 2 + hi) * 16 + m];
    acc = __builtin_amdgcn_wmma_f32_16x16x4_f32(false, a, false, b,
                                                (short)0, acc, false, false);
  }
#pragma unroll
  for (int v = 0; v < 8; ++v)
    out[pbase + (long)(ht * 16 + v + hi * 8) * WW + wt * 16 + m] = acc[v];
}

// ---- Pass 3: blur along C (len 3, radius 4 => fixed 3x3 mix), float4 stream
__global__ void k_cblur(const v4f* __restrict__ in, v4f* __restrict__ out) {
  int idx = blockIdx.x * blockDim.x + threadIdx.x;   // 0 .. NB*SC4
  if (idx >= NB * SC4) return;
  const int  n = idx >> 16;            // /65536 (= SC4)
  const int  r = idx & (SC4 - 1);
  const long base = (long)n * SN4 + r;
  v4f v0 = in[base], v1 = in[base + SC4], v2 = in[base + 2 * SC4];
#pragma unroll
  for (int c = 0; c < 3; ++c) {
    float m0 = 0.f, m1 = 0.f, m2 = 0.f;
#pragma unroll
    for (int t = 0; t < 9; ++t) {
      int s = c - 4 + t;
      s = ((s % 6) + 6) % 6; s = (s < 3) ? s : 5 - s;  // symmetric fold, period 6
      float kv = gk(t);                                 // compile-time constant
      m0 += (s == 0) ? kv : 0.f;
      m1 += (s == 1) ? kv : 0.f;
      m2 += (s == 2) ? kv : 0.f;
    }
    out[base + (long)c * SC4] = m0 * v0 + m1 * v1 + m2 * v2;
  }
}

// ---- Pass 4: blur along N (len 16), float4 stream, all 16 entries in regs
__global__ void k_nblur(const v4f* __restrict__ in, v4f* __restrict__ out) {
  int idx = blockIdx.x * blockDim.x + threadIdx.x;    // 0 .. SN4
  if (idx >= SN4) return;
  v4f x[16];
#pragma unroll
  for (int n = 0; n < 16; ++n) x[n] = in[(long)n * SN4 + idx];
#pragma unroll
  for (int n = 0; n < 16; ++n) {
    v4f acc = {};
#pragma unroll
    for (int t = 0; t < 9; ++t) {
      int s = n - 4 + t;
      s = (s < 0) ? (-1 - s) : (s > 15 ? 31 - s : s);  // symmetric fold
      acc += gk(t) * x[s];                             // constants after unroll
    }
    out[(long)n * SN4 + idx] = acc;
  }
}

// ---- Loss: R = (I+1e-6)*exp(-S), L = exp(S); deterministic 2-stage reduction
__global__ void k_reduce(const v4f* __restrict__ Ie, const v4f* __restrict__ Ig,
                         const v4f* __restrict__ Se, const v4f* __restrict__ Sg,
                         float* __restrict__ part) {
  __shared__ float sm[256];
  float acc = 0.f;
  for (long i = (long)blockIdx.x * blockDim.x + threadIdx.x; i < NEL / 4;
       i += (long)gridDim.x * blockDim.x) {
    v4f ie = Ie[i], ig = Ig[i], se = Se[i], sg = Sg[i];
#pragma unroll
    for (int c = 0; c < 4; ++c) {
      float le = __expf(se[c]), lg = __expf(sg[c]);
      float re = (ie[c] + 1e-6f) * __expf(-se[c]);
      float rg = (ig[c] + 1e-6f) * __expf(-sg[c]);
      float dr = re - rg, dl = le - lg;
      acc += dr * dr + dl * dl;
    }
  }
  sm[threadIdx.x] = acc;
  __syncthreads();
  for (int s = 128; s > 0; s >>= 1) {
    if (threadIdx.x < s) sm[threadIdx.x] += sm[threadIdx.x + s];
    __syncthreads();
  }
  if (threadIdx.x == 0) part[blockIdx.x] = sm[0];
}

__global__ void k_final(const float* __restrict__ part, float* __restrict__ out) {
  __shared__ float sm[256];
  sm[threadIdx.x] = part[threadIdx.x] + part[threadIdx.x + 256] +
                    part[threadIdx.x + 512] + part[threadIdx.x + 768];
  __syncthreads();
  for (int s = 128; s > 0; s >>= 1) {
    if (threadIdx.x < s) sm[threadIdx.x] += sm[threadIdx.x + s];
    __syncthreads();
  }
  if (threadIdx.x == 0) out[0] = sm[0] / (float)NEL;
}

extern "C" void kernel_launch(void* const* d_in, const int* in_sizes, int n_in,
                              void* d_out, int out_size, void* d_ws, size_t ws_size,
                              hipStream_t stream) {
  const float* Ie = (const float*)d_in[0];
  const float* Ig = (const float*)d_in[1];
  float* Se   = (float*)d_ws;        // smooth(enhanced)
  float* Sg   = Se + NEL;            // smooth(gt)
  float* P    = Sg + NEL;            // shared ping buffer
  float* part = P + NEL;             // 1024 block partials
  dim3 blk(256);

  // enhanced: in -> P(W,log) -> Se(H) -> P(C) -> Se(N)
  k_wblur_log<<<6144, blk, 0, stream>>>(Ie, P);
  k_hblur    <<<6144, blk, 0, stream>>>(P, Se);
  k_cblur    <<<NB * SC4 / 256, blk, 0, stream>>>((const v4f*)Se, (v4f*)P);
  k_nblur    <<<SN4 / 256, blk, 0, stream>>>((const v4f*)P, (v4f*)Se);
  // gt
  k_wblur_log<<<6144, blk, 0, stream>>>(Ig, P);
  k_hblur    <<<6144, blk, 0, stream>>>(P, Sg);
  k_cblur    <<<NB * SC4 / 256, blk, 0, stream>>>((const v4f*)Sg, (v4f*)P);
  k_nblur    <<<SN4 / 256, blk, 0, stream>>>((const v4f*)P, (v4f*)Sg);

  k_reduce<<<1024, blk, 0, stream>>>((const v4f*)Ie, (const v4f*)Ig,
                                     (const v4f*)Se, (const v4f*)Sg, part);
  k_final <<<1, blk, 0, stream>>>(part, (float*)d_out);
}